// MultiHeadEMA_26731876450905
// MI455X (gfx1250) — compile-verified
//
#include <hip/hip_runtime.h>
#include <math.h>

// MI455X / gfx1250: wave32, WMMA f32 16x16x4, 320KB LDS per WGP.
// MultiHeadEMA (L=4096,B=4,E=1024,N=16) as a chunked low-rank SSM scan:
//   out = T@X (intra-chunk triangular Toeplitz, WMMA)
//       + Bmat@H (rank-16 cross-chunk states, WMMA)
//   states H from S = A@X (WMMA) + 64-step scalar scan.
// One workgroup per channel e; both directions fused; residual+SiLU epilogue.

typedef __attribute__((ext_vector_type(2))) float v2f;
typedef __attribute__((ext_vector_type(8))) float v8f;

#define L_SEQ   4096
#define BATCH   4
#define EMB     1024
#define NDIM16  16
#define CHUNK   64
#define NCHUNK  64              // L_SEQ / CHUNK
#define COLS    256             // NCHUNK * BATCH (columns = (chunk,batch) pairs)
#define XS_STR  65              // 64 local cols + 1 pad (bank-conflict-free)
#define SS_STR  257             // 256 cols + 1 pad

__global__ __launch_bounds__(128) void ema_chunked_wmma_kernel(
    const float* __restrict__ x,      // (L,B,E)
    const float* __restrict__ alpha,  // (2E,16,1)
    const float* __restrict__ delta,  // (2E,16,1)
    const float* __restrict__ beta,   // (2E,16,1)
    const float* __restrict__ gamma,  // (2E,16)
    const float* __restrict__ omega,  // (E,)
    float* __restrict__ out)          // (L,B,E)
{
    __shared__ float Xs[CHUNK * XS_STR];        // 16,640 B  quarter of X columns
    __shared__ float Ss[2][NDIM16 * SS_STR];    // 32,896 B  chunk sums -> states (in-place scan)
    __shared__ float pw[2][NDIM16][CHUNK];      //  8,192 B  q^0..q^63 per state
    __shared__ float kv[2][CHUNK];              //    512 B  kernel taps k[0..63]
    __shared__ float ctab[2][NDIM16];
    __shared__ float qtab[2][NDIM16];
    __shared__ float qCtab[2][NDIM16];          // q^64

    const int e    = blockIdx.x;
    const int tid  = threadIdx.x;
    const int lane = tid & 31;
    const int wv   = tid >> 5;        // wave id 0..3
    const int hi   = lane >> 4;       // lane half (ISA f32 WMMA layout)
    const int lr   = lane & 15;

    // ---- Phase A: per-state parameters and power tables -------------------
    if (tid < 32) {
        const int dir = tid >> 4;
        const int n   = tid & 15;
        const int idx = (e + dir * EMB) * NDIM16 + n;
        const float a  = 1.0f / (1.0f + expf(-alpha[idx]));
        const float dd = 1.0f / (1.0f + expf(-delta[idx]));
        const float q  = 1.0f - a * dd;                  // in (0,1)
        const float c  = a * beta[idx] * gamma[idx] * 0.25f; // scale = sqrt(1/16)
        ctab[dir][n] = c;
        qtab[dir][n] = q;
        float v = 1.0f;
        for (int l = 0; l < CHUNK; ++l) { pw[dir][n][l] = v; v *= q; }
        qCtab[dir][n] = v;                               // q^CHUNK
    }
    __syncthreads();

    // ---- Phase B: kernel taps k[dir][l] = sum_n c_n q_n^l -----------------
    {
        const int dir = tid >> 6;
        const int l   = tid & 63;
        float s = 0.0f;
        #pragma unroll
        for (int n = 0; n < NDIM16; ++n) s += ctab[dir][n] * pw[dir][n][l];
        kv[dir][l] = s;
    }
    __syncthreads();

    // ---- Phase C: chunk sums  S[dir] = A[dir](16x64) @ X(64x256) ----------
    // A fwd[n][ss] = q^(63-ss), A bwd[n][ss] = q^ss. Column quarters through LDS.
    for (int qtr = 0; qtr < 4; ++qtr) {
        for (int t = tid; t < CHUNK * 64; t += 128) {
            const int ss   = t >> 6;
            const int c    = t & 63;
            const int colg = (qtr << 6) + c;         // global column = chunk*4+b
            const int ci   = colg >> 2;
            const int b    = colg & 3;
            const int l    = (ci << 6) + ss;
            Xs[ss * XS_STR + c] = x[(size_t)l * (BATCH * EMB) + b * EMB + e];
        }
        __syncthreads();

        const int colL = (wv << 4) + lr;             // wave's local column tile
        v8f s0 = {}, s1 = {};
        for (int k0 = 0; k0 < CHUNK; k0 += 4) {
            const int kk = k0 + 2 * hi;              // f32 A/B frag: K = 2*hi + j
            v2f bx; bx.x = Xs[kk * XS_STR + colL];
                    bx.y = Xs[(kk + 1) * XS_STR + colL];
            v2f a0; a0.x = pw[0][lr][63 - kk];  a0.y = pw[0][lr][62 - kk];
            v2f a1; a1.x = pw[1][lr][kk];       a1.y = pw[1][lr][kk + 1];
            s0 = __builtin_amdgcn_wmma_f32_16x16x4_f32(false, a0, false, bx, (short)0, s0, false, false);
            s1 = __builtin_amdgcn_wmma_f32_16x16x4_f32(false, a1, false, bx, (short)0, s1, false, false);
        }
        const int colG = (qtr << 6) + colL;
        #pragma unroll
        for (int r = 0; r < 8; ++r) {                // C/D layout: M = r + 8*hi
            const int row = r + 8 * hi;
            Ss[0][row * SS_STR + colG] = s0[r];
            Ss[1][row * SS_STR + colG] = s1[r];
        }
        __syncthreads();
    }

    // ---- Phase D: 64-step chunk scan, in place on Ss ----------------------
    // fwd: H[i] = q^C H[i-1] + S[i-1]   (store state entering chunk i)
    // bwd: G[i] = S[i] + q^C G[i+1]     (store G[i+1] for chunk i)
    {
        const int dir = tid >> 6;                    // wave-uniform
        const int n   = (tid >> 2) & 15;
        const int b   = tid & 3;
        const float qC = qCtab[dir][n];
        float h = 0.0f;
        if (dir == 0) {
            for (int i = 0; i < NCHUNK; ++i) {
                const int idx = n * SS_STR + (i << 2) + b;
                const float s = Ss[0][idx];
                Ss[0][idx] = h;
                h = qC * h + s;
            }
        } else {
            for (int i = NCHUNK - 1; i >= 0; --i) {
                const int idx = n * SS_STR + (i << 2) + b;
                const float s = Ss[1][idx];
                Ss[1][idx] = h;
                h = s + qC * h;
            }
        }
    }
    __syncthreads();

    // ---- Phase E: Out = T@X + Bmat@H, fused residual + SiLU ---------------
    const float om = omega[e];
    for (int qtr = 0; qtr < 4; ++qtr) {
        for (int t = tid; t < CHUNK * 64; t += 128) {
            const int ss   = t >> 6;
            const int c    = t & 63;
            const int colg = (qtr << 6) + c;
            const int ci   = colg >> 2;
            const int b    = colg & 3;
            const int l    = (ci << 6) + ss;
            Xs[ss * XS_STR + c] = x[(size_t)l * (BATCH * EMB) + b * EMB + e];
        }
        __syncthreads();

        const int colL = (wv << 4) + lr;
        const int colG = (qtr << 6) + colL;
        v8f acc[4] = {};                             // 4 M-tiles (tt = 0..63)

        // Intra-chunk Toeplitz: T1 (causal) and T2 (anti-causal), synthesized
        // from the taps in LDS -- never materialized.
        for (int k0 = 0; k0 < CHUNK; k0 += 4) {
            const int kk = k0 + 2 * hi;
            v2f bx; bx.x = Xs[kk * XS_STR + colL];
                    bx.y = Xs[(kk + 1) * XS_STR + colL];
            #pragma unroll
            for (int mt = 0; mt < 4; ++mt) {
                const int tt = (mt << 4) + lr;       // output row (A frag M)
                v2f aT;                              // T1[tt][ss] = k1[tt-ss], ss<=tt
                aT.x = (kk     <= tt) ? kv[0][tt - kk]     : 0.0f;
                aT.y = (kk + 1 <= tt) ? kv[0][tt - kk - 1] : 0.0f;
                acc[mt] = __builtin_amdgcn_wmma_f32_16x16x4_f32(false, aT, false, bx, (short)0, acc[mt], false, false);
                v2f aU;                              // T2[tt][ss] = k2[ss-tt-1], ss>tt
                aU.x = (kk     > tt) ? kv[1][kk - tt - 1] : 0.0f;
                aU.y = (kk + 1 > tt) ? kv[1][kk - tt]     : 0.0f;
                acc[mt] = __builtin_amdgcn_wmma_f32_16x16x4_f32(false, aU, false, bx, (short)0, acc[mt], false, false);
            }
        }
        // Cross-chunk rank-16 term: Bmat(64x16) @ H(16x256)
        for (int k0 = 0; k0 < NDIM16; k0 += 4) {
            const int kk = k0 + 2 * hi;
            v2f bh0; bh0.x = Ss[0][kk * SS_STR + colG]; bh0.y = Ss[0][(kk + 1) * SS_STR + colG];
            v2f bh1; bh1.x = Ss[1][kk * SS_STR + colG]; bh1.y = Ss[1][(kk + 1) * SS_STR + colG];
            #pragma unroll
            for (int mt = 0; mt < 4; ++mt) {
                const int tt = (mt << 4) + lr;
                v2f aB0;                             // c_n * q^(tt+1)
                aB0.x = ctab[0][kk]     * qtab[0][kk]     * pw[0][kk][tt];
                aB0.y = ctab[0][kk + 1] * qtab[0][kk + 1] * pw[0][kk + 1][tt];
                acc[mt] = __builtin_amdgcn_wmma_f32_16x16x4_f32(false, aB0, false, bh0, (short)0, acc[mt], false, false);
                v2f aB1;                             // c_n * q^(63-tt)
                aB1.x = ctab[1][kk]     * pw[1][kk][63 - tt];
                aB1.y = ctab[1][kk + 1] * pw[1][kk + 1][63 - tt];
                acc[mt] = __builtin_amdgcn_wmma_f32_16x16x4_f32(false, aB1, false, bh1, (short)0, acc[mt], false, false);
            }
        }
        // Epilogue: residual + SiLU, direct store (x re-read hits 192MB L2).
        #pragma unroll
        for (int mt = 0; mt < 4; ++mt) {
            #pragma unroll
            for (int r = 0; r < 8; ++r) {
                const int tt = (mt << 4) + r + 8 * hi;
                const int ci = colG >> 2;
                const int b  = colG & 3;
                const int l  = (ci << 6) + tt;
                const size_t g = (size_t)l * (BATCH * EMB) + b * EMB + e;
                const float v = acc[mt][r] + x[g] * om;
                out[g] = v / (1.0f + expf(-v));
            }
        }
        __syncthreads();
    }
}

extern "C" void kernel_launch(void* const* d_in, const int* in_sizes, int n_in,
                              void* d_out, int out_size, void* d_ws, size_t ws_size,
                              hipStream_t stream) {
    (void)in_sizes; (void)n_in; (void)out_size; (void)d_ws; (void)ws_size;
    const float* x     = (const float*)d_in[0];
    const float* alpha = (const float*)d_in[1];
    const float* delta = (const float*)d_in[2];
    const float* beta  = (const float*)d_in[3];
    const float* gamma = (const float*)d_in[4];
    const float* omega = (const float*)d_in[5];
    float* out = (float*)d_out;
    hipLaunchKernelGGL(ema_chunked_wmma_kernel, dim3(EMB), dim3(128), 0, stream,
                       x, alpha, delta, beta, gamma, omega, out);
}